// Renderer_88699664597866
// MI455X (gfx1250) — compile-verified
//
#include <hip/hip_runtime.h>
#include <stdint.h>

#define IMG       128
#define NPTS      16384
#define NB        2
#define FUV       140.0f     // 35 * 128 / 32
#define DIST_CAM  1.2f
#define R_SIL     0.008f
#define R_DEP     0.012f
#define ZSENT     1e30f
#define CHUNK_PTS 1024                    // points staged per LDS chunk
#define NCHUNK    (NPTS / CHUNK_PTS)      // 16
#define PT_PER_THR (CHUNK_PTS / 256)      // 4 float4 loads per thread per chunk

typedef __attribute__((ext_vector_type(2))) float v2f;
typedef __attribute__((ext_vector_type(8))) float v8f;
typedef __attribute__((ext_vector_type(4))) int   v4i;

// ---------------- CDNA5 async global->LDS copy (ASYNCcnt path) ----------------
#if __has_builtin(__builtin_amdgcn_global_load_async_to_lds_b128)
#define HAVE_ASYNC_BUILTIN 1
#else
#define HAVE_ASYNC_BUILTIN 0
#endif

__device__ __forceinline__ void async_copy16(const float4* g, float4* l) {
#if HAVE_ASYNC_BUILTIN
  __builtin_amdgcn_global_load_async_to_lds_b128((v4i*)g, (v4i*)l, 0, 0);
#else
  const uint64_t ga = (uint64_t)(uintptr_t)g;
  const uint32_t la = (uint32_t)(uintptr_t)l;
  asm volatile("global_load_async_to_lds_b128 %0, %1, off"
               :: "v"(la), "v"(ga) : "memory");
#endif
}

__device__ __forceinline__ void wait_async0() {
#if __has_builtin(__builtin_amdgcn_s_wait_asynccnt)
  __builtin_amdgcn_s_wait_asynccnt(0);
#else
  asm volatile("s_wait_asynccnt 0x0" ::: "memory");
#endif
}

// ---------------- Threefry-2x32-20 (JAX PRNG) ----------------
__device__ __forceinline__ void tf2x32(uint32_t k0, uint32_t k1,
                                       uint32_t x0, uint32_t x1,
                                       uint32_t &o0, uint32_t &o1) {
  const uint32_t ks2 = k0 ^ k1 ^ 0x1BD11BDAu;
  uint32_t a = x0 + k0, b = x1 + k1;
#define TFR(r) { a += b; b = (b << (r)) | (b >> (32 - (r))); b ^= a; }
  TFR(13) TFR(15) TFR(26) TFR(6)  a += k1;  b += ks2 + 1u;
  TFR(17) TFR(29) TFR(16) TFR(24) a += ks2; b += k0  + 2u;
  TFR(13) TFR(15) TFR(26) TFR(6)  a += k0;  b += k1  + 3u;
  TFR(17) TFR(29) TFR(16) TFR(24) a += k1;  b += ks2 + 4u;
  TFR(13) TFR(15) TFR(26) TFR(6)  a += ks2; b += k0  + 5u;
#undef TFR
  o0 = a; o1 = b;
}

__device__ __forceinline__ float bits_to_unit(uint32_t bits) {
  return __uint_as_float((bits >> 9) | 0x3f800000u) - 1.0f;
}

__device__ __forceinline__ void rand_cam(int b, float azim0,
                                         float &elev_d, float &azim_d) {
  uint32_t e02a, e02b, e13a, e13b;
  tf2x32(0u, 42u, 0u, 2u, e02a, e02b);   // split: pairs (0,2) / (1,3) of iota(4)
  tf2x32(0u, 42u, 1u, 3u, e13a, e13b);
  uint32_t we0, we1, wa0, wa1;
  tf2x32(e02a, e13a, 0u, 1u, we0, we1);  // uniform bits for ke
  tf2x32(e02b, e13b, 0u, 1u, wa0, wa1);  // uniform bits for ka
  const float ue = bits_to_unit(b == 0 ? we0 : we1);
  const float ua = bits_to_unit(b == 0 ? wa0 : wa1);
  elev_d = ue * 60.0f - 30.0f;
  azim_d = ua * 300.0f + azim0 + 30.0f;
}

// ---------------- camera ----------------
__device__ void look_at(float elev_deg, float azim_deg, float R[3][3], float T[3]) {
  const float d2r = 0.017453292519943295f;
  const float e = elev_deg * d2r, az = azim_deg * d2r;
  const float ce = cosf(e), se = sinf(e), sa = sinf(az), ca = cosf(az);
  const float cx = DIST_CAM * ce * sa, cy = DIST_CAM * se, cz = DIST_CAM * ce * ca;
  const float inv = 1.0f / (sqrtf(cx*cx + cy*cy + cz*cz) + 1e-8f);
  const float zx = -cx * inv, zy = -cy * inv, zz = -cz * inv;
  float xx = zz, xy = 0.0f, xz = -zx;                 // up x z, up=(0,1,0)
  const float xn = 1.0f / (sqrtf(xx*xx + xy*xy + xz*xz) + 1e-8f);
  xx *= xn; xy *= xn; xz *= xn;
  const float yx = zy*xz - zz*xy, yy = zz*xx - zx*xz, yz = zx*xy - zy*xx;
  R[0][0]=xx; R[1][0]=xy; R[2][0]=xz;
  R[0][1]=yx; R[1][1]=yy; R[2][1]=yz;
  R[0][2]=zx; R[1][2]=zy; R[2][2]=zz;
  T[0] = -(cx*R[0][0] + cy*R[1][0] + cz*R[2][0]);
  T[1] = -(cx*R[0][1] + cy*R[1][1] + cz*R[2][1]);
  T[2] = -(cx*R[0][2] + cy*R[1][2] + cz*R[2][2]);
}

__device__ __forceinline__ float4 project_pt(float px, float py, float pz,
                                             const float R[3][3], const float T[3]) {
  const float X0 = px*R[0][0] + py*R[1][0] + pz*R[2][0] + T[0];
  const float X1 = px*R[0][1] + py*R[1][1] + pz*R[2][1] + T[1];
  const float X2 = px*R[0][2] + py*R[1][2] + pz*R[2][2] + T[2];
  const float zs = (X2 > 1e-5f) ? X2 : 1.0f;
  const float xn = 1.0f - (FUV * X0 / zs + 64.0f) * (1.0f/64.0f);
  const float yn = 1.0f - (FUV * X1 / zs + 64.0f) * (1.0f/64.0f);
  return make_float4(-2.0f*xn, -2.0f*yn, xn*xn + yn*yn, X2);  // A features + depth
}

// ---------------- K1: cameras + projection ----------------
__global__ void prep_kernel(const float* __restrict__ pc, const float* __restrict__ param,
                            float4* __restrict__ pts0, float4* __restrict__ ptsd,
                            unsigned int* __restrict__ area) {
  const int gid = blockIdx.x * blockDim.x + threadIdx.x;
  if (gid < NB) area[gid] = 0u;                       // re-zero every call
  if (gid >= NB * NPTS) return;
  const int b = gid / NPTS;
  const float elev0 = param[b*24 + 3];                // param[b,1,0]
  const float azim0 = param[b*24 + 0];                // param[b,0,0]
  float R0[3][3], T0[3], Rd[3][3], Td[3];
  look_at(elev0, azim0, R0, T0);
  float ed, ad; rand_cam(b, azim0, ed, ad);
  look_at(ed, ad, Rd, Td);
  const float* p = pc + (size_t)gid * 6;
  const float px = p[0], py = p[1], pz = p[2];
  pts0[gid] = project_pt(px, py, pz, R0, T0);
  ptsd[gid] = project_pt(px, py, pz, Rd, Td);
}

// ---------------- WMMA d^2 tile: 16 points (M) x 16 pixels (N), K=4 ----------------
__device__ __forceinline__ v8f tile_d2(v2f a, v2f b) {
  v8f c = {0.f,0.f,0.f,0.f,0.f,0.f,0.f,0.f};
  return __builtin_amdgcn_wmma_f32_16x16x4_f32(false, a, false, b,
                                               (short)0, c, false, false);
}

__device__ __forceinline__ void insert8(float z, float d, float topz[8], float topd[8]) {
  float cz = z, cd = d;
#pragma unroll
  for (int k = 0; k < 8; ++k) {
    const bool s = cz < topz[k];
    const float tz = topz[k], td = topd[k];
    if (s) { topz[k] = cz; topd[k] = cd; cz = tz; cd = td; }
  }
}

// ---------------- unified raster kernel ----------------
// MODE 0: silhouette top-8 + alpha composite -> out0[b*P+p]
// MODE 1: any-hit -> atomicAdd(area[b])
// MODE 2: adaptive top-8 -> out0 (depth, 8/px), out1 (dist, 8/px)
template <int MODE>
__global__ void __launch_bounds__(256) raster_kernel(const float4* __restrict__ pts,
                                                     unsigned int* area,
                                                     float* __restrict__ out0,
                                                     float* __restrict__ out1) {
  __shared__ float4 sbuf[2][CHUNK_PTS];               // 32 KB double buffer
  const int tid  = threadIdx.x;
  const int lane = tid & 31;
  const int wave = blockIdx.x * 8 + (tid >> 5);
  const int b  = wave >> 10;                          // whole block shares b (8 | 1024)
  const int pg = wave & 1023;
  const int n  = lane & 15;
  const int p  = pg * 16 + n;
  const int ix = p & (IMG - 1), iy = p >> 7;
  const float gx = 1.0f - (2.0f*ix + 1.0f) * (1.0f/IMG);
  const float gy = 1.0f - (2.0f*iy + 1.0f) * (1.0f/IMG);
  const bool hi = lane >= 16;
  v2f bm; bm.x = hi ? (gx*gx + gy*gy) : gx;           // B rows K0,K1 | K2,K3
  bm.y = hi ? 1.0f : gy;
  float r2;
  if (MODE == 0)      r2 = R_SIL * R_SIL;
  else if (MODE == 1) r2 = R_DEP * R_DEP;
  else { const float pr = (float)area[b] * (0.057f/16384.0f); r2 = pr * pr; }
  const int h8 = (lane >> 4) << 3;

  float topz[8], topd[8];
  bool hit = false;
#pragma unroll
  for (int i = 0; i < 8; ++i) { topz[i] = ZSENT; topd[i] = -1.0f; }

  const float4* gbase = pts + (size_t)b * NPTS;

  // stage chunk 0
#pragma unroll
  for (int k = 0; k < PT_PER_THR; ++k)
    async_copy16(&gbase[tid + k*256], &sbuf[0][tid + k*256]);
  wait_async0();
  __syncthreads();

  for (int c = 0; c < NCHUNK; ++c) {
    const int cur = c & 1;
    if (c + 1 < NCHUNK) {                             // prefetch next chunk (async)
      const float4* src = gbase + (c + 1) * CHUNK_PTS;
#pragma unroll
      for (int k = 0; k < PT_PER_THR; ++k)
        async_copy16(&src[tid + k*256], &sbuf[cur ^ 1][tid + k*256]);
    }
#pragma unroll 2
    for (int t = 0; t < CHUNK_PTS/16; ++t) {
      const float4 pd = sbuf[cur][t*16 + n];
      v2f am; am.x = hi ? 1.0f : pd.x; am.y = hi ? pd.z : pd.y;  // A K0,K1 | K2,K3
      const v8f d2 = tile_d2(am, bm);
      bool pre = false;
#pragma unroll
      for (int i = 0; i < 8; ++i) pre = pre || (d2[i] < r2);
      if (__any(pre)) {
#pragma unroll
        for (int i = 0; i < 8; ++i) {
          const float z  = __shfl(pd.w, h8 + i, 32);
          const float dd = d2[i];
          if (dd < r2 && z > 0.0f) {
            if (MODE == 1) hit = true;
            else           insert8(z, dd, topz, topd);
          }
        }
      }
    }
    wait_async0();                                     // our async stores landed
    __syncthreads();                                   // everyone done with sbuf[cur]
  }

  if (MODE == 1) {
    const bool oh = __shfl((int)hit, (lane & 15) + 16, 32) != 0;
    const bool hitAll = hit || oh;
    const unsigned long long m = __ballot(hitAll);
    if (lane == 0) atomicAdd(&area[b], (unsigned int)__popcll(m & 0xFFFFull));
    return;
  }

  // merge the two lane-halves' partial top-8 lists into lanes 0..15
#pragma unroll
  for (int i = 0; i < 8; ++i) {
    const float oz = __shfl(topz[i], (lane & 15) + 16, 32);
    const float od = __shfl(topd[i], (lane & 15) + 16, 32);
    insert8(oz, od, topz, topd);
  }

  if (lane < 16) {
    if (MODE == 0) {
      float trans = 1.0f, acc = 0.0f;
#pragma unroll
      for (int i = 0; i < 8; ++i) {
        const bool valid = topz[i] < 0.5f * ZSENT;
        const float a = valid ? (1.0f - topd[i] / r2) : 0.0f;
        acc += a * trans;
        trans *= (1.0f - a);
      }
      out0[(size_t)b * (IMG*IMG) + p] = acc * 255.0f;
    } else {
      const size_t pix = (size_t)b * (IMG*IMG) + p;
#pragma unroll
      for (int i = 0; i < 8; ++i) {
        const bool valid = topz[i] < 0.5f * ZSENT;
        out0[pix * 8 + i] = valid ? topz[i] : -1.0f;
        out1[pix * 8 + i] = valid ? topd[i] : -1.0f;
      }
    }
  }
}

// ---------------- launcher ----------------
extern "C" void kernel_launch(void* const* d_in, const int* in_sizes, int n_in,
                              void* d_out, int out_size, void* d_ws, size_t ws_size,
                              hipStream_t stream) {
  const float* pc    = (const float*)d_in[0];   // (2,16384,6)
  const float* param = (const float*)d_in[1];   // (2,8,3)
  float* out = (float*)d_out;                   // sihlt | depth_img | dist_top

  float4* pts0 = (float4*)d_ws;
  float4* ptsd = pts0 + (size_t)NB * NPTS;
  unsigned int* area = (unsigned int*)(ptsd + (size_t)NB * NPTS);

  prep_kernel<<<(NB*NPTS + 255)/256, 256, 0, stream>>>(pc, param, pts0, ptsd, area);

  float* depth_out = out + (size_t)NB * IMG * IMG;           // +32768
  float* dist_out  = depth_out + (size_t)NB * IMG * IMG * 8; // +262144

  // 2048 waves per raster pass: 8 waves/block -> 256 blocks
  raster_kernel<0><<<256, 256, 0, stream>>>(pts0, area, out, nullptr);
  raster_kernel<1><<<256, 256, 0, stream>>>(ptsd, area, nullptr, nullptr);
  raster_kernel<2><<<256, 256, 0, stream>>>(ptsd, area, depth_out, dist_out);
}